// Attention_32298154066539
// MI455X (gfx1250) — compile-verified
//
#include <hip/hip_runtime.h>
#include <math.h>

// Problem constants (match reference)
#define B_  8
#define S_  2048
#define D_  512
#define H_  8
#define DK_ 64

typedef __attribute__((ext_vector_type(2))) float v2f;
typedef __attribute__((ext_vector_type(8))) float v8f;

// D = A(16x4 f32) * B(4x16 f32) + C(16x16 f32)  -- native fp32 matrix pipe
__device__ __forceinline__ v8f wmma4(v2f a, v2f b, v8f c) {
    return __builtin_amdgcn_wmma_f32_16x16x4_f32(
        /*neg_a=*/false, a, /*neg_b=*/false, b,
        /*c_mod=*/(short)0, c, /*reuse_a=*/false, /*reuse_b=*/false);
}

// -------------------------------------------------------------------------
// Kernel 1: T5 relative position bias -> d_out tail, layout [H, S, S]
// grid (S_/256, S_), block 256: blockIdx.y = q, thread -> k
// -------------------------------------------------------------------------
__global__ void __launch_bounds__(256) bias_kernel(
        const float* __restrict__ bias_emb, float* __restrict__ out_bias) {
    const int k = blockIdx.x * 256 + threadIdx.x;
    const int q = blockIdx.y;
    const int rp = k - q;
    int rb = (rp > 0) ? 16 : 0;           // nb = NUM_BUCKETS/2 = 16
    const int rpa = (rp < 0) ? -rp : rp;
    int add;
    if (rpa < 8) {                         // ne = nb/2 = 8 exact buckets
        add = rpa;
    } else {
        // 8 + int( log(rpa/8)/log(128/8) * 8 ), clamped to 15
        float lr = __logf((float)rpa * 0.125f) * (1.0f / __logf(16.0f));
        int lsp = 8 + (int)(lr * 8.0f);
        add = (lsp < 15) ? lsp : 15;
    }
    rb += add;
    #pragma unroll
    for (int h = 0; h < H_; ++h) {
        out_bias[((size_t)h * S_ + q) * S_ + k] = bias_emb[rb * H_ + h];
    }
}

// -------------------------------------------------------------------------
// Kernel 2/4: Y[16384,512] = X[16384,512] @ W[512,512], f32 WMMA 16x16x4.
// Block = 256 thr = 8 waves; wave w -> rows m0..m0+15, cols n0..n0+63.
// bhsd_layout==1 scatters Y into [B,H,S,DK] (for Q/K/V), else row-major.
// grid = (512/64, 16384/128) = (8, 128)
// -------------------------------------------------------------------------
__global__ void __launch_bounds__(256, 1) gemm512_kernel(
        const float* __restrict__ X, const float* __restrict__ W,
        float* __restrict__ Y, int bhsd_layout) {
    const int lane = threadIdx.x & 31;
    const int wave = threadIdx.x >> 5;
    const int mr   = lane & 15;            // M (A) / N (B,C) index within tile
    const int g    = lane >> 4;            // half-wave selector (K pair / row half)
    const int m0   = blockIdx.y * 128 + wave * 16;
    const int n0   = blockIdx.x * 64;

    const float* xrow = X + (size_t)(m0 + mr) * D_;
    v8f acc0 = {}, acc1 = {}, acc2 = {}, acc3 = {};

    for (int k0 = 0; k0 < D_; k0 += 4) {
        // A fragment: v0 = X[m][k0+2g], v1 = X[m][k0+2g+1]  (8B aligned)
        v2f a = *(const v2f*)(xrow + k0 + 2 * g);
        const float* wr0 = W + (size_t)(k0 + 2 * g) * 512 + n0 + mr;
        const float* wr1 = wr0 + 512;
        v2f b;
        b.x = wr0[0];  b.y = wr1[0];  acc0 = wmma4(a, b, acc0);
        b.x = wr0[16]; b.y = wr1[16]; acc1 = wmma4(a, b, acc1);
        b.x = wr0[32]; b.y = wr1[32]; acc2 = wmma4(a, b, acc2);
        b.x = wr0[48]; b.y = wr1[48]; acc3 = wmma4(a, b, acc3);
    }

    // C/D layout: vgpr r, lanes 0-15 -> row r, lanes 16-31 -> row 8+r, col = mr
    #pragma unroll
    for (int t = 0; t < 4; ++t) {
        v8f acc = (t == 0) ? acc0 : (t == 1) ? acc1 : (t == 2) ? acc2 : acc3;
        const int col = n0 + 16 * t + mr;
        #pragma unroll
        for (int r = 0; r < 8; ++r) {
            const int row = m0 + r + 8 * g;
            size_t idx;
            if (bhsd_layout) {
                const int bb = row >> 11, ss = row & (S_ - 1);
                const int hh = col >> 6,  dd = col & (DK_ - 1);
                idx = (((size_t)bb * H_ + hh) * S_ + ss) * DK_ + dd;
            } else {
                idx = (size_t)row * D_ + col;
            }
            Y[idx] = acc[r];
        }
    }
}

// -------------------------------------------------------------------------
// Kernel 3: flash attention per (b,h). Block = 8 waves; wave -> 16 q-rows.
// scores = Q K^T + bias + mask ; online softmax ; O = P V.
// grid = (S_/128, H_, B_)
// -------------------------------------------------------------------------
__global__ void __launch_bounds__(256, 1) attn_kernel(
        const float* __restrict__ q_ws, const float* __restrict__ k_ws,
        const float* __restrict__ v_ws, const float* __restrict__ bias,
        const float* __restrict__ mask, float* __restrict__ attn_out) {
    __shared__ float ptile[8][16][17];     // padded: conflict-free transpose

    const int lane = threadIdx.x & 31;
    const int wave = threadIdx.x >> 5;
    const int mr   = lane & 15;
    const int g    = lane >> 4;
    const int q0   = blockIdx.x * 128 + wave * 16;
    const int h    = blockIdx.y;
    const int b    = blockIdx.z;
    const size_t bh = (size_t)b * H_ + h;

    // Preload Q fragments for all 16 k-steps of DK=64 (lives in regs)
    const float* qrow = q_ws + (bh * S_ + q0 + mr) * DK_;
    v2f aq[16];
    #pragma unroll
    for (int kk = 0; kk < 16; ++kk)
        aq[kk] = *(const v2f*)(qrow + 4 * kk + 2 * g);

    v8f o0 = {}, o1 = {}, o2 = {}, o3 = {};
    float mrow[8], lrow[8];
    #pragma unroll
    for (int r = 0; r < 8; ++r) { mrow[r] = -INFINITY; lrow[r] = 0.f; }

    const float* bias_h = bias + (size_t)h * S_ * S_;
    const float* mask_b = mask + (size_t)b * S_;

    for (int j0 = 0; j0 < S_; j0 += 16) {
        // ---- S = Q K^T : B fragment v0 = K[j0+n][4kk+2g]
        const float* krow = k_ws + (bh * S_ + j0 + mr) * DK_;
        v8f s = {};
        #pragma unroll
        for (int kk = 0; kk < 16; ++kk) {
            v2f bk = *(const v2f*)(krow + 4 * kk + 2 * g);
            s = wmma4(aq[kk], bk, s);
        }

        // ---- + position bias + mask (C layout: row = r+8g, col = mr)
        const float mval = mask_b[j0 + mr];
        #pragma unroll
        for (int r = 0; r < 8; ++r) {
            const int qg = q0 + r + 8 * g;
            s[r] += bias_h[(size_t)qg * S_ + j0 + mr] + mval;
        }

        // ---- online softmax (row reductions across the 16 lanes of each half)
        #pragma unroll
        for (int r = 0; r < 8; ++r) {
            float t = s[r];
            t = fmaxf(t, __shfl_xor(t, 1, 32));
            t = fmaxf(t, __shfl_xor(t, 2, 32));
            t = fmaxf(t, __shfl_xor(t, 4, 32));
            t = fmaxf(t, __shfl_xor(t, 8, 32));
            const float mn    = fmaxf(mrow[r], t);
            const float scale = __expf(mrow[r] - mn);
            mrow[r] = mn;
            const float p = __expf(s[r] - mn);
            s[r] = p;
            float rs = p;
            rs += __shfl_xor(rs, 1, 32);
            rs += __shfl_xor(rs, 2, 32);
            rs += __shfl_xor(rs, 4, 32);
            rs += __shfl_xor(rs, 8, 32);
            lrow[r] = lrow[r] * scale + rs;
            o0[r] *= scale; o1[r] *= scale; o2[r] *= scale; o3[r] *= scale;
        }

        // ---- transpose P: C layout -> A-fragment layout via LDS (same-wave,
        //      DS ops are in-order; explicit dscnt wait before consuming)
        #pragma unroll
        for (int r = 0; r < 8; ++r)
            ptile[wave][r + 8 * g][mr] = s[r];
        __asm__ volatile("s_wait_dscnt 0" ::: "memory");
        v2f ap[4];
        #pragma unroll
        for (int kk = 0; kk < 4; ++kk) {
            ap[kk].x = ptile[wave][mr][4 * kk + 2 * g];
            ap[kk].y = ptile[wave][mr][4 * kk + 2 * g + 1];
        }

        // ---- O += P V : B fragment v0 = V[j0+4kk+2g][n]
        const float* vbase = v_ws + (bh * S_ + j0) * DK_ + mr;
        #pragma unroll
        for (int kk = 0; kk < 4; ++kk) {
            const float* vr0 = vbase + (size_t)(4 * kk + 2 * g) * DK_;
            v2f bv;
            bv.x = vr0[0];  bv.y = vr0[DK_ + 0];  o0 = wmma4(ap[kk], bv, o0);
            bv.x = vr0[16]; bv.y = vr0[DK_ + 16]; o1 = wmma4(ap[kk], bv, o1);
            bv.x = vr0[32]; bv.y = vr0[DK_ + 32]; o2 = wmma4(ap[kk], bv, o2);
            bv.x = vr0[48]; bv.y = vr0[DK_ + 48]; o3 = wmma4(ap[kk], bv, o3);
        }
    }

    // ---- normalize and store to [B, S, H*DK] for the output projection
    float* outp = attn_out + ((size_t)b * S_ + q0) * (H_ * DK_) + h * DK_;
    #pragma unroll
    for (int r = 0; r < 8; ++r) {
        const float inv = 1.0f / lrow[r];
        float* orow = outp + (size_t)(r + 8 * g) * (H_ * DK_);
        orow[mr +  0] = o0[r] * inv;
        orow[mr + 16] = o1[r] * inv;
        orow[mr + 32] = o2[r] * inv;
        orow[mr + 48] = o3[r] * inv;
    }
}

// -------------------------------------------------------------------------
// Host-side launch
// inputs: 0=hidden_states 1=mask 2=Wq 3=Wk 4=Wv 5=Wo 6=bias_emb
// d_out = [out (B*S*D) | position_bias (H*S*S)]
// -------------------------------------------------------------------------
extern "C" void kernel_launch(void* const* d_in, const int* in_sizes, int n_in,
                              void* d_out, int out_size, void* d_ws, size_t ws_size,
                              hipStream_t stream) {
    const float* X        = (const float*)d_in[0];
    const float* mask     = (const float*)d_in[1];
    const float* Wq       = (const float*)d_in[2];
    const float* Wk       = (const float*)d_in[3];
    const float* Wv       = (const float*)d_in[4];
    const float* Wo       = (const float*)d_in[5];
    const float* bias_emb = (const float*)d_in[6];

    float* out      = (float*)d_out;
    float* out_bias = out + (size_t)B_ * S_ * D_;      // 8,388,608 floats in

    const size_t qkv_elems = (size_t)B_ * H_ * S_ * DK_;  // 8,388,608
    float* q_ws = (float*)d_ws;
    float* k_ws = q_ws + qkv_elems;
    float* v_ws = k_ws + qkv_elems;
    float* a_ws = v_ws + qkv_elems;                     // [B,S,H*DK]

    // 1) position bias straight into d_out tail (also consumed by attention)
    bias_kernel<<<dim3(S_ / 256, S_), 256, 0, stream>>>(bias_emb, out_bias);

    // 2) Q/K/V projections (scatter to [B,H,S,DK])
    dim3 ggrid(D_ / 64, (B_ * S_) / 128);
    gemm512_kernel<<<ggrid, 256, 0, stream>>>(X, Wq, q_ws, 1);
    gemm512_kernel<<<ggrid, 256, 0, stream>>>(X, Wk, k_ws, 1);
    gemm512_kernel<<<ggrid, 256, 0, stream>>>(X, Wv, v_ws, 1);

    // 3) flash attention with online softmax
    attn_kernel<<<dim3(S_ / 128, H_, B_), 256, 0, stream>>>(
        q_ws, k_ws, v_ws, out_bias, mask, a_ws);

    // 4) output projection -> d_out head (row-major [B,S,D])
    gemm512_kernel<<<ggrid, 256, 0, stream>>>(a_ws, Wo, out, 0);
}